// CenterNetBBoxModule_65137474011606
// MI455X (gfx1250) — compile-verified
//
#include <hip/hip_runtime.h>
#include <math.h>

// Problem geometry from the reference's setup_inputs(): n=16, C=1, H=W=128, nobj=128.
#define NOBJ 128
#define HDIM 128
#define WDIM 128

typedef __attribute__((address_space(3))) char lds_char_t;

__global__ __launch_bounds__(256) void centernet_gt_heatmap_kernel(
    const float* __restrict__ bboxes,   // [n, NOBJ, 5]
    float* __restrict__ out,            // [n, 1, H, W]
    int n)
{
    __shared__ alignas(16) float sBBox[NOBJ * 5];   // 2560 B staging of this batch's bbox slice
    __shared__ alignas(16) int4  sParams[NOBJ];     // {cxi, cyi, ri_or_-1, bits(-1/(2*sigma^2))}

    const int tid   = threadIdx.x;
    const int batch = blockIdx.y;
    const int tile  = blockIdx.x;       // 16 tiles of 8 rows each

    // ---------------- async global -> LDS copy of bbox slice (CDNA5 path) ----------------
    // 2560 bytes = 160 lanes x b128. Per-lane LDS address in a VGPR, global address in a
    // VGPR pair; tracked by ASYNCcnt, drained with s_wait_asynccnt before the barrier.
    const float* gsrc = bboxes + (size_t)batch * (NOBJ * 5);
    if (tid < (NOBJ * 5 / 4)) {
        unsigned ldsOff =
            (unsigned)(unsigned long long)(lds_char_t*)(&sBBox[tid * 4]);
        unsigned long long gaddr = (unsigned long long)(gsrc + tid * 4);
        asm volatile("global_load_async_to_lds_b128 %0, %1, off"
                     :: "v"(ldsOff), "v"(gaddr)
                     : "memory");
    }
    asm volatile("s_wait_asynccnt 0" ::: "memory");
    __syncthreads();

    // ---------------- per-object parameters (threads 0..127) ----------------
    if (tid < NOBJ) {
        const float* bb = &sBBox[tid * 5];
        float x1 = bb[0] * (float)WDIM;
        float y1 = bb[1] * (float)HDIM;
        float x2 = bb[2] * (float)WDIM;
        float y2 = bb[3] * (float)HDIM;
        float vf = bb[4];
        float w  = x2 - x1;
        float h  = y2 - y1;

        // gaussian_radius(min_overlap = 0.7), op-for-op in fp32:
        float b1  = h + w;
        float c1  = w * h * 0.3f / 1.7f;            // w*h*(1-mo)/(1+mo)
        float sq1 = sqrtf(b1 * b1 - 4.0f * c1);
        float r1  = (b1 + sq1) * 0.5f;

        float b2  = 2.0f * (h + w);
        float c2  = 0.3f * w * h;                   // (1-mo)*w*h
        float sq2 = sqrtf(b2 * b2 - 16.0f * c2);    // b2^2 - 4*a2*c2, a2=4
        float r2  = (b2 + sq2) * 0.5f;

        float b3  = -1.4f * (h + w);                // -2*mo*(h+w)
        float c3  = -0.3f * w * h;                  // (mo-1)*w*h
        float sq3 = sqrtf(b3 * b3 - 11.2f * c3);    // b3^2 - 4*(4*mo)*c3
        float r3  = (b3 + sq3) * 0.5f;

        float r = fminf(fminf(r1, r2), r3);
        // jnp.minimum propagates NaN; nan_to_num maps it to 0 (fminf would not).
        if (r1 != r1 || r2 != r2 || r3 != r3) r = 0.0f;
        r = fmaxf(r, 0.0f);

        bool valid = (vf == 1.0f) && (w > 0.0f) && (h > 0.0f);
        float cx = (x1 + x2) * 0.5f;
        float cy = (y1 + y2) * 0.5f;
        int cxi = (int)cx;                          // trunc toward zero, like .astype(int32)
        int cyi = (int)cy;
        int ri  = (int)r;

        float sig  = (float)(2 * ri + 1) / 6.0f;
        float ninv = -1.0f / (2.0f * sig * sig);

        int4 q;
        q.x = cxi;
        q.y = cyi;
        q.z = valid ? ri : -1;                      // ri=-1 => inside test always false
        q.w = __float_as_int(ninv);
        sParams[tid] = q;
    }
    __syncthreads();

    // ---------------- pixel loop: one wave per row, 4 px per lane ----------------
    const int row = tile * 8 + (tid >> 5);          // wave-uniform row
    const int x0  = (tid & 31) << 2;                // 4 consecutive pixels

    // max over objects of the (masked) exp argument; exp applied once at the end.
    float m[4];
    m[0] = -INFINITY; m[1] = -INFINITY; m[2] = -INFINITY; m[3] = -INFINITY;

    #pragma unroll 4
    for (int o = 0; o < NOBJ; ++o) {
        int4 p = sParams[o];                        // ds_load_b128, wave-broadcast
        float ninv = __int_as_float(p.w);
        int dy  = row - p.y;
        int ady = (dy < 0) ? -dy : dy;
        int dy2 = dy * dy;
        int okY = (ady <= p.z);
        int dx0 = x0 - p.x;
        #pragma unroll
        for (int j = 0; j < 4; ++j) {
            int dx  = dx0 + j;
            int adx = (dx < 0) ? -dx : dx;
            bool in = okY && (adx <= p.z);
            float a = (float)(dx * dx + dy2) * ninv;
            m[j] = fmaxf(m[j], in ? a : -INFINITY);
        }
    }

    float4 res;
    res.x = expf(m[0]);                             // expf(-inf) == 0 -> "no object" case
    res.y = expf(m[1]);
    res.z = expf(m[2]);
    res.w = expf(m[3]);

    float* dst = out + (size_t)batch * (HDIM * WDIM) + row * WDIM + x0;
    *(float4*)dst = res;                            // coalesced 512 B per wave
}

extern "C" void kernel_launch(void* const* d_in, const int* in_sizes, int n_in,
                              void* d_out, int out_size, void* d_ws, size_t ws_size,
                              hipStream_t stream) {
    // d_in[0] = hm   [n, 1, 128, 128] (shape/dtype only)
    // d_in[1] = bboxes [n, 128, 5]
    const float* bboxes = (const float*)d_in[1];
    float* out = (float*)d_out;
    const int n = in_sizes[1] / (NOBJ * 5);         // 16

    dim3 grid(HDIM / 8, n);                         // 16 row-tiles x n batches
    centernet_gt_heatmap_kernel<<<grid, 256, 0, stream>>>(bboxes, out, n);
}